// GATE_83090437309062
// MI455X (gfx1250) — compile-verified
//
#include <hip/hip_runtime.h>
#include <math.h>

typedef __attribute__((ext_vector_type(2))) float v2f;
typedef __attribute__((ext_vector_type(8))) float v8f;

#define ALPHA_C 0.8f
#define WD_C    1e-4f

// ---------------------------------------------------------------------------
// fp32 WMMA GEMM: C[M x Nc] = A[M x K] @ B[K x Nc]
//   transB=0: B(k,n) = Bsrc[k*Nc + n]
//   transB=1: B(k,n) = Bsrc[n*ldbSrc + k]   (i.e. C = A @ Bsrc^T)
// Block = 256 threads (8 waves). Each block owns one 32-column slab of B
// staged in LDS (padded stride K+4 -> conflict-free ds_load_b64). Each wave
// computes a 16x32 C tile: two v8f accumulators sharing one A fetch, so A
// global traffic is halved vs one tile/wave and every A float2 feeds two
// back-to-back v_wmma_f32_16x16x4_f32.
// ---------------------------------------------------------------------------
__global__ __launch_bounds__(256) void gemm_wmma_f32(
    const float* __restrict__ A, const float* __restrict__ Bsrc,
    float* __restrict__ C, int M, int K, int Nc, int transB, int ldbSrc)
{
    __shared__ float ldsB[32 * 260];            // 32 cols x (K<=256 + 4 pad)
    const int tid  = threadIdx.x;
    const int nt   = blockIdx.x;                // 32-col slab index
    const int ldsS = K + 4;

    // cooperative, coalesced load of B slab (columns nt*32 .. nt*32+31)
    const int total = K * 32;
    if (transB) {
        for (int idx = tid; idx < total; idx += 256) {
            int n = idx / K, k = idx - n * K;   // consecutive tid -> consecutive k
            ldsB[n * ldsS + k] = Bsrc[(size_t)(nt * 32 + n) * ldbSrc + k];
        }
    } else {
        for (int idx = tid; idx < total; idx += 256) {
            int k = idx >> 5, n = idx & 31;     // consecutive tid -> consecutive n
            ldsB[n * ldsS + k] = Bsrc[(size_t)k * Nc + nt * 32 + n];
        }
    }
    __syncthreads();

    const int wave = tid >> 5;
    const int lane = tid & 31;
    const int half = lane >> 4;                 // K-pair selector (ISA A layout)
    const int m    = lane & 15;                 // row within tile / col within tile
    const int mt   = blockIdx.y * 8 + wave;
    const int mtiles = (M + 15) >> 4;
    if (mt >= mtiles) return;                   // wave-uniform: EXEC all-1s below

    int arow = mt * 16 + m; if (arow >= M) arow = M - 1;
    const float* __restrict__ Ar  = A + (size_t)arow * K;
    const float* __restrict__ Bl0 = &ldsB[m * ldsS];
    const float* __restrict__ Bl1 = &ldsB[(m + 16) * ldsS];

    v8f acc0 = {};
    v8f acc1 = {};
    for (int k = 0; k < K; k += 4) {
        v2f a, b0, b1;
        // A 16x4 layout: lanes0-15 hold K=k,k+1 ; lanes16-31 hold K=k+2,k+3
        a.x = Ar[k + 2 * half];
        a.y = Ar[k + 2 * half + 1];
        // B 4x16 layout mirrors A with N across lanes
        b0 = *(const v2f*)(Bl0 + k + 2 * half);
        b1 = *(const v2f*)(Bl1 + k + 2 * half);
        acc0 = __builtin_amdgcn_wmma_f32_16x16x4_f32(false, a, false, b0,
                                                     (short)0, acc0, false, false);
        acc1 = __builtin_amdgcn_wmma_f32_16x16x4_f32(false, a, false, b1,
                                                     (short)0, acc1, false, false);
    }

    const int col0 = nt * 32 + m;
    #pragma unroll
    for (int j = 0; j < 8; ++j) {               // VGPR j: M=j (lo lanes), M=j+8 (hi)
        int row = mt * 16 + 8 * half + j;
        if (row < M) {
            C[(size_t)row * Nc + col0]      = acc0[j];
            C[(size_t)row * Nc + col0 + 16] = acc1[j];
        }
    }
}

// ---------------------------------------------------------------------------
// fused 4-way GEMV: f = H(row) . v for v00,v01,pv00,pv01 ; one wave per node
// ---------------------------------------------------------------------------
__global__ __launch_bounds__(256) void attn_feats(
    const float* __restrict__ H,
    const float* __restrict__ v00, const float* __restrict__ v01,
    const float* __restrict__ pv00, const float* __restrict__ pv01,
    float* __restrict__ f1, float* __restrict__ f2,
    float* __restrict__ pf1, float* __restrict__ pf2, int n)
{
    int gw   = (int)((blockIdx.x * 256u + threadIdx.x) >> 5);
    int lane = threadIdx.x & 31;
    if (gw >= n) return;
    const float4 h = *(const float4*)(H + (size_t)gw * 128 + lane * 4);
    const float4 a = *(const float4*)(v00  + lane * 4);
    const float4 b = *(const float4*)(v01  + lane * 4);
    const float4 c = *(const float4*)(pv00 + lane * 4);
    const float4 d = *(const float4*)(pv01 + lane * 4);
    float s0 = h.x*a.x + h.y*a.y + h.z*a.z + h.w*a.w;
    float s1 = h.x*b.x + h.y*b.y + h.z*b.z + h.w*b.w;
    float s2 = h.x*c.x + h.y*c.y + h.z*c.z + h.w*c.w;
    float s3 = h.x*d.x + h.y*d.y + h.z*d.z + h.w*d.w;
    #pragma unroll
    for (int o = 16; o > 0; o >>= 1) {
        s0 += __shfl_xor(s0, o, 32);
        s1 += __shfl_xor(s1, o, 32);
        s2 += __shfl_xor(s2, o, 32);
        s3 += __shfl_xor(s3, o, 32);
    }
    if (lane == 0) { f1[gw] = s0; f2[gw] = s1; pf1[gw] = s2; pf2[gw] = s3; }
}

// ---------------------------------------------------------------------------
// edge softmax passes
// ---------------------------------------------------------------------------
__device__ __forceinline__ void atomicMaxF(float* addr, float val)
{
    int* ai = (int*)addr;
    if (val >= 0.0f) atomicMax(ai, __float_as_int(val));
    else             atomicMin((unsigned int*)ai, (unsigned int)__float_as_int(val));
}

__global__ __launch_bounds__(256) void edge_logit_max(
    const int* __restrict__ rows, const int* __restrict__ cols,
    const float* __restrict__ f1, const float* __restrict__ f2,
    float* __restrict__ elog, float* __restrict__ rmax, int E)
{
    int e = blockIdx.x * 256 + threadIdx.x;
    if (e >= E) return;
    float l = f1[rows[e]] + f2[cols[e]];
    elog[e] = l;
    atomicMaxF(rmax + rows[e], l);
}

__global__ __launch_bounds__(256) void edge_exp_sum(
    const int* __restrict__ rows, float* __restrict__ elog,
    const float* __restrict__ rmax, float* __restrict__ rsum, int E)
{
    int e = blockIdx.x * 256 + threadIdx.x;
    if (e >= E) return;
    float v = expf(elog[e] - rmax[rows[e]]);
    elog[e] = v;
    atomicAdd(rsum + rows[e], v);
}

__global__ __launch_bounds__(256) void edge_norm(
    const int* __restrict__ rows, float* __restrict__ elog,
    const float* __restrict__ rsum, float scale, int E)
{
    int e = blockIdx.x * 256 + threadIdx.x;
    if (e >= E) return;
    elog[e] = scale * elog[e] / rsum[rows[e]];
}

// ---------------------------------------------------------------------------
// SpMM scatter: out[rows[e]] += vals[e] * H[cols[e]]  (dim 128)
// 4 edges per wave; prefetch next edge's gather row (global_prefetch_b8) so
// the next 512B of H is in flight while this edge's f32 atomics issue.
// ---------------------------------------------------------------------------
#define EPW 4
__global__ __launch_bounds__(256) void spmm_scatter(
    const int* __restrict__ rows, const int* __restrict__ cols,
    const float* __restrict__ vals, const float* __restrict__ H,
    float* __restrict__ out, int E)
{
    int w    = (int)((blockIdx.x * 256u + threadIdx.x) >> 5);
    int lane = threadIdx.x & 31;
    int e0   = w * EPW;
    #pragma unroll
    for (int i = 0; i < EPW; ++i) {
        int e = e0 + i;
        if (e >= E) return;
        if (e + 1 < E)
            __builtin_prefetch(H + (size_t)cols[e + 1] * 128 + lane * 4, 0, 0);
        int r = rows[e], c = cols[e];
        float v = vals[e];
        const float4 h = *(const float4*)(H + (size_t)c * 128 + lane * 4);
        float* o = out + (size_t)r * 128 + lane * 4;
        atomicAdd(o + 0, v * h.x);
        atomicAdd(o + 1, v * h.y);
        atomicAdd(o + 2, v * h.z);
        atomicAdd(o + 3, v * h.w);
    }
}

// ---------------------------------------------------------------------------
// elementwise / init / reduction
// ---------------------------------------------------------------------------
__global__ __launch_bounds__(256) void elu_inplace(float* __restrict__ x, int n)
{
    int i = blockIdx.x * 256 + threadIdx.x;
    if (i >= n) return;
    float v = x[i];
    x[i] = v > 0.0f ? v : (expf(v) - 1.0f);
}

__global__ __launch_bounds__(256) void fill_f32(float* __restrict__ p, float v, int n)
{
    int i = blockIdx.x * 256 + threadIdx.x;
    if (i < n) p[i] = v;
}

__global__ __launch_bounds__(256) void reduce_sqdiff(
    const float* __restrict__ a, const float* __restrict__ b,
    float* __restrict__ acc, int n)
{
    __shared__ float sdata[256];
    float s = 0.0f;
    for (size_t i = blockIdx.x * 256u + threadIdx.x; i < (size_t)n;
         i += (size_t)gridDim.x * 256u) {
        float d = a[i] - b[i];
        s += d * d;
    }
    sdata[threadIdx.x] = s; __syncthreads();
    for (int o = 128; o > 0; o >>= 1) {
        if ((int)threadIdx.x < o) sdata[threadIdx.x] += sdata[threadIdx.x + o];
        __syncthreads();
    }
    if (threadIdx.x == 0) atomicAdd(acc, sdata[0]);
}

__global__ __launch_bounds__(256) void reduce_sq(
    const float* __restrict__ a, float* __restrict__ acc, int n)
{
    __shared__ float sdata[256];
    float s = 0.0f;
    for (size_t i = blockIdx.x * 256u + threadIdx.x; i < (size_t)n;
         i += (size_t)gridDim.x * 256u)
        s += a[i] * a[i];
    sdata[threadIdx.x] = s; __syncthreads();
    for (int o = 128; o > 0; o >>= 1) {
        if ((int)threadIdx.x < o) sdata[threadIdx.x] += sdata[threadIdx.x + o];
        __syncthreads();
    }
    if (threadIdx.x == 0) atomicAdd(acc, sdata[0]);
}

__global__ void finalize_loss(const float* __restrict__ scal, float* __restrict__ out)
{
    out[0] = sqrtf(scal[0]) + WD_C * scal[1];
}

// ---------------------------------------------------------------------------
// host orchestration
// ---------------------------------------------------------------------------
extern "C" void kernel_launch(void* const* d_in, const int* in_sizes, int n_in,
                              void* d_out, int out_size, void* d_ws, size_t ws_size,
                              hipStream_t stream)
{
    const float* X    = (const float*)d_in[0];
    const float* W0   = (const float*)d_in[1];   // 256 x 128
    const float* W1   = (const float*)d_in[2];   // 128 x 64
    const float* v00  = (const float*)d_in[3];
    const float* v01  = (const float*)d_in[4];
    const float* pv00 = (const float*)d_in[5];
    const float* pv01 = (const float*)d_in[6];
    const int* rows   = (const int*)d_in[7];
    const int* cols   = (const int*)d_in[8];
    const int* prows  = (const int*)d_in[9];
    const int* pcols  = (const int*)d_in[10];

    const int N  = in_sizes[0] / 256;
    const int E  = in_sizes[7];
    const int Ep = in_sizes[9];

    float* out  = (float*)d_out;
    float* loss = out;
    float* Henc = out + 1;                         // N x 64
    float* Xrec = out + 1 + (size_t)N * 64;        // N x 256

    float* ws   = (float*)d_ws;
    float* bufA  = ws;  ws += (size_t)N * 128;     // H1 / D
    float* bufB  = ws;  ws += (size_t)N * 128;     // H1a acc / Dacc
    float* f1    = ws;  ws += N;
    float* f2    = ws;  ws += N;
    float* pf1   = ws;  ws += N;
    float* pf2   = ws;  ws += N;
    float* rmax1 = ws;  ws += N;                   // rmax1/rmax2 contiguous
    float* rmax2 = ws;  ws += N;
    float* rsum1 = ws;  ws += N;                   // rsum1/rsum2 contiguous
    float* rsum2 = ws;  ws += N;
    float* ec    = ws;  ws += E;
    float* epc   = ws;  ws += Ep;
    float* scal  = ws;  ws += 8;

    const int mtiles = (N + 15) / 16;
    const dim3 gemmGridN128(128 / 32, (mtiles + 7) / 8);
    const dim3 gemmGridN64 ( 64 / 32, (mtiles + 7) / 8);
    const dim3 gemmGridN256(256 / 32, (mtiles + 7) / 8);
    const int eBlk  = (E  + 255) / 256;
    const int epBlk = (Ep + 255) / 256;
    const int eWav  = ((E  + EPW - 1) / EPW * 32 + 255) / 256;   // EPW edges/wave
    const int epWav = ((Ep + EPW - 1) / EPW * 32 + 255) / 256;
    const int nWav  = (N * 32 + 255) / 256;

    // deterministic re-init of all accumulators (graph replay safe)
    hipMemsetAsync(rsum1, 0, (size_t)2 * N * sizeof(float), stream);
    hipMemsetAsync(scal, 0, 8 * sizeof(float), stream);
    fill_f32<<<(2 * N + 255) / 256, 256, 0, stream>>>(rmax1, -__builtin_huge_valf(), 2 * N);

    // ---- encoder ----
    // H1 = X @ W0
    gemm_wmma_f32<<<gemmGridN128, 256, 0, stream>>>(X, W0, bufA, N, 256, 128, 0, 0);
    // attention features
    attn_feats<<<nWav, 256, 0, stream>>>(bufA, v00, v01, pv00, pv01, f1, f2, pf1, pf2, N);
    // edge softmax (graph + prune graph), scaled by (1-alpha)/alpha
    edge_logit_max<<<eBlk, 256, 0, stream>>>(rows, cols, f1, f2, ec, rmax1, E);
    edge_logit_max<<<epBlk, 256, 0, stream>>>(prows, pcols, pf1, pf2, epc, rmax2, Ep);
    edge_exp_sum<<<eBlk, 256, 0, stream>>>(rows, ec, rmax1, rsum1, E);
    edge_exp_sum<<<epBlk, 256, 0, stream>>>(prows, epc, rmax2, rsum2, Ep);
    edge_norm<<<eBlk, 256, 0, stream>>>(rows, ec, rsum1, 1.0f - ALPHA_C, E);
    edge_norm<<<epBlk, 256, 0, stream>>>(prows, epc, rsum2, ALPHA_C, Ep);
    // H1a = (1-a)*spmm(c,H1) + a*spmm(pc,H1) ; elu
    hipMemsetAsync(bufB, 0, (size_t)N * 128 * sizeof(float), stream);
    spmm_scatter<<<eWav, 256, 0, stream>>>(rows, cols, ec, bufA, bufB, E);
    spmm_scatter<<<epWav, 256, 0, stream>>>(prows, pcols, epc, bufA, bufB, Ep);
    elu_inplace<<<(N * 128 + 255) / 256, 256, 0, stream>>>(bufB, N * 128);
    // H_enc = H1a @ W1
    gemm_wmma_f32<<<gemmGridN64, 256, 0, stream>>>(bufB, W1, Henc, N, 128, 64, 0, 0);

    // ---- decoder ----
    // D = H_enc @ W1^T
    gemm_wmma_f32<<<gemmGridN128, 256, 0, stream>>>(Henc, W1, bufA, N, 64, 128, 1, 64);
    // D = (1-a)*spmm(c,D) + a*spmm(pc,D) ; elu
    hipMemsetAsync(bufB, 0, (size_t)N * 128 * sizeof(float), stream);
    spmm_scatter<<<eWav, 256, 0, stream>>>(rows, cols, ec, bufA, bufB, E);
    spmm_scatter<<<epWav, 256, 0, stream>>>(prows, pcols, epc, bufA, bufB, Ep);
    elu_inplace<<<(N * 128 + 255) / 256, 256, 0, stream>>>(bufB, N * 128);
    // X_ = D @ W0^T
    gemm_wmma_f32<<<gemmGridN256, 256, 0, stream>>>(bufB, W0, Xrec, N, 128, 256, 1, 128);

    // ---- loss ----
    reduce_sqdiff<<<1024, 256, 0, stream>>>(X, Xrec, scal + 0, N * 256);
    reduce_sq<<<64, 256, 0, stream>>>(W0, scal + 1, 256 * 128);
    reduce_sq<<<64, 256, 0, stream>>>(W1, scal + 1, 128 * 64);
    finalize_loss<<<1, 1, 0, stream>>>(scal, loss);
}